// LocalRefineStage_84413287236173
// MI455X (gfx1250) — compile-verified
//
#include <hip/hip_runtime.h>
#include <hip/hip_bf16.h>

// ---------------------------------------------------------------------------
// LocalRefineStage for MI455X (gfx1250, wave32).
//   Kernel 1: KNN (k=8) over LDS-staged partial cloud + mean -> MLP input.
//   Kernel 2: fused 4-layer MLP using fp32 WMMA (v_wmma_f32_16x16x4_f32),
//             2 row-tiles per wave so every B-fragment load feeds 2 WMMAs.
// ---------------------------------------------------------------------------

typedef __attribute__((ext_vector_type(2))) float v2f;
typedef __attribute__((ext_vector_type(8))) float v8f;

#define BB   8
#define MM   2048
#define NN   8192
#define KNN  8
#define HID  128
#define INP  6
#define INP8 8      // padded MLP input width (cols 6,7 = 0)
#define OUTW 2

// ---------------------------------------------------------------------------
// Kernel 1: per query point, find 8 nearest partial points, emit
//   x[row][0..2] = predicted, x[row][3..5] = mean(neigh), x[row][6..7] = 0
// ---------------------------------------------------------------------------
__global__ void __launch_bounds__(256)
knn_seed_kernel(const float* __restrict__ partial,
                const float* __restrict__ predicted,
                float* __restrict__ xout) {
  __shared__ float sp[MM * 3];                 // 24 KB: partial cloud of batch b
  const int b   = blockIdx.y;
  const int tid = threadIdx.x;

  for (int i = tid; i < MM * 3; i += 256)
    sp[i] = partial[b * MM * 3 + i];
  __syncthreads();

  const int n = blockIdx.x * 256 + tid;
  const int row = b * NN + n;
  const float px = predicted[row * 3 + 0];
  const float py = predicted[row * 3 + 1];
  const float pz = predicted[row * 3 + 2];

  // Fully unrolled register-resident sorted top-8 (ascending distance).
  float bd[KNN]; int bi[KNN];
#pragma unroll
  for (int j = 0; j < KNN; ++j) { bd[j] = 3.4e38f; bi[j] = 0; }

  for (int m = 0; m < MM; ++m) {
    const float dx = px - sp[3 * m + 0];
    const float dy = py - sp[3 * m + 1];
    const float dz = pz - sp[3 * m + 2];
    float d = dx * dx + dy * dy + dz * dz;
    if (d < bd[KNN - 1]) {
      int id = m;
#pragma unroll
      for (int j = 0; j < KNN; ++j) {          // stable insertion (strict <)
        const bool sw = d < bd[j];
        const float td = bd[j]; const int ti = bi[j];
        bd[j] = sw ? d : td;  bi[j] = sw ? id : ti;
        d     = sw ? td : d;  id    = sw ? ti : id;
      }
    }
  }

  float sx = 0.f, sy = 0.f, sz = 0.f;
#pragma unroll
  for (int j = 0; j < KNN; ++j) {              // same summation order as top_k
    const int m = bi[j];
    sx += sp[3 * m + 0]; sy += sp[3 * m + 1]; sz += sp[3 * m + 2];
  }

  float* xr = xout + (size_t)row * INP8;
  xr[0] = px;          xr[1] = py;          xr[2] = pz;
  xr[3] = sx * 0.125f; xr[4] = sy * 0.125f; xr[5] = sz * 0.125f;
  xr[6] = 0.f;         xr[7] = 0.f;
}

// ---------------------------------------------------------------------------
// fp32 WMMA helper. A: 16x4 (lane L: row = L&15, VGPR pair holds K =
// (L>>4)*2 + {0,1}).  B: 4x16 (lane L: col = L&15, VGPR pair holds K =
// (L>>4)*2 + {0,1}).  C/D: VGPR j = rows j (lanes 0-15) / j+8 (lanes 16-31),
// col = lane&15 within a 16-wide N tile.
// ---------------------------------------------------------------------------
__device__ __forceinline__ v8f wmma_f32(v2f a, v2f b, v8f c) {
  return __builtin_amdgcn_wmma_f32_16x16x4_f32(
      /*neg_a=*/false, a, /*neg_b=*/false, b,
      /*c_mod=*/(short)0, c, /*reuse_a=*/false, /*reuse_b=*/false);
}

__device__ __forceinline__ float relu(float x) { return x > 0.f ? x : 0.f; }

// ---------------------------------------------------------------------------
// Kernel 2: fused MLP. 4 waves/block, 32 rows/wave (two 16-row A tiles that
// share every B fragment). LDS: 3 staging tiles of 32x128 f32 per wave
// (h, seed, ht) = 192 KB/block.
// ---------------------------------------------------------------------------
__global__ void __launch_bounds__(128)
mlp_refine_kernel(const float* __restrict__ xin,
                  const float* __restrict__ predicted,
                  const float* __restrict__ W1, const float* __restrict__ b1,
                  const float* __restrict__ W2, const float* __restrict__ b2,
                  const float* __restrict__ Wd, const float* __restrict__ bd,
                  const float* __restrict__ Wc, const float* __restrict__ bc,
                  float* __restrict__ out) {
  extern __shared__ float smem[];
  const int tid   = threadIdx.x;
  const int wave  = tid >> 5;
  const int lane  = tid & 31;
  const int lrow  = lane & 15;          // A-row / B-col / D-col within tile
  const int khalf = lane >> 4;          // which K pair this lane supplies

  float* lds_h    = smem + wave * (32 * HID);                   // h tiles
  float* lds_seed = smem + 4 * (32 * HID) + wave * (32 * HID);  // seed tiles
  float* lds_ht   = smem + 8 * (32 * HID) + wave * (32 * HID);  // ht tiles

  const int rowBase = blockIdx.x * 128 + wave * 32;             // 32 rows/wave
  const int dRow = (lane < 16) ? 0 : 8;  // D layout row offset for this lane

  // ---- Layer 1: h = relu(X @ W1^T + b1), X is 32x8 (padded) --------------
  {
    const float* xr0 = xin + (size_t)(rowBase + lrow) * INP8;
    const float* xr1 = xin + (size_t)(rowBase + 16 + lrow) * INP8;
    const float kmask = khalf ? 0.f : 1.f;   // A cols 6,7 are zero anyway
    v2f a00, a01, a10, a11;
    a00.x = xr0[khalf * 2 + 0];     a00.y = xr0[khalf * 2 + 1];     // K 0..3
    a01.x = xr0[4 + khalf * 2 + 0]; a01.y = xr0[4 + khalf * 2 + 1]; // K 4..7
    a10.x = xr1[khalf * 2 + 0];     a10.y = xr1[khalf * 2 + 1];
    a11.x = xr1[4 + khalf * 2 + 0]; a11.y = xr1[4 + khalf * 2 + 1];
#pragma unroll
    for (int nt = 0; nt < 8; ++nt) {
      const int n = nt * 16 + lrow;
      const float bv = b1[n];
      v8f c0, c1;
#pragma unroll
      for (int j = 0; j < 8; ++j) { c0[j] = bv; c1[j] = bv; }    // bias in C
      v2f bf0, bf1;
      const int k0 = khalf * 2;                                  // 0 or 2
      bf0.x = W1[n * INP + k0];        bf0.y = W1[n * INP + k0 + 1];
      bf1.x = kmask * W1[n * INP + 4]; bf1.y = kmask * W1[n * INP + 5];
      c0 = wmma_f32(a00, bf0, c0);  c0 = wmma_f32(a01, bf1, c0);
      c1 = wmma_f32(a10, bf0, c1);  c1 = wmma_f32(a11, bf1, c1);
#pragma unroll
      for (int j = 0; j < 8; ++j) {
        const int r = j + dRow;
        lds_h[r * HID + n]        = relu(c0[j]);
        lds_h[(16 + r) * HID + n] = relu(c1[j]);
      }
    }
  }

  // ---- Layer 2: seed = h @ W2^T + b2 -------------------------------------
#pragma unroll
  for (int nt = 0; nt < 8; ++nt) {
    const int n = nt * 16 + lrow;
    const float bv = b2[n];
    v8f c0, c1;
#pragma unroll
    for (int j = 0; j < 8; ++j) { c0[j] = bv; c1[j] = bv; }
    const float* w2r = W2 + n * HID;                  // B[k][n] = W2[n][k]
    for (int k4 = 0; k4 < HID; k4 += 4) {
      const int kb = k4 + khalf * 2;
      v2f bf, aA, aB;
      bf.x = w2r[kb];                       bf.y = w2r[kb + 1];
      aA.x = lds_h[lrow * HID + kb];        aA.y = lds_h[lrow * HID + kb + 1];
      aB.x = lds_h[(16 + lrow) * HID + kb]; aB.y = lds_h[(16 + lrow) * HID + kb + 1];
      c0 = wmma_f32(aA, bf, c0);
      c1 = wmma_f32(aB, bf, c1);
    }
#pragma unroll
    for (int j = 0; j < 8; ++j) {
      const int r = j + dRow;
      lds_seed[r * HID + n]        = c0[j];           // no activation
      lds_seed[(16 + r) * HID + n] = c1[j];
    }
  }

  // ---- Layer 3+4 per deconv tap t: ht = relu(seed @ Wd[:,:,t] + bd),
  //      out = ht @ Wc^T + bc, child = predicted + out ----------------------
  for (int t = 0; t < OUTW; ++t) {
#pragma unroll
    for (int nt = 0; nt < 8; ++nt) {
      const int n = nt * 16 + lrow;
      const float bv = bd[n];
      v8f c0, c1;
#pragma unroll
      for (int j = 0; j < 8; ++j) { c0[j] = bv; c1[j] = bv; }
      for (int k4 = 0; k4 < HID; k4 += 4) {
        const int kb = k4 + khalf * 2;
        v2f bf, aA, aB;
        bf.x = Wd[(kb * HID + n) * OUTW + t];         // B[k][n] = Wd[k][n][t]
        bf.y = Wd[((kb + 1) * HID + n) * OUTW + t];
        aA.x = lds_seed[lrow * HID + kb];
        aA.y = lds_seed[lrow * HID + kb + 1];
        aB.x = lds_seed[(16 + lrow) * HID + kb];
        aB.y = lds_seed[(16 + lrow) * HID + kb + 1];
        c0 = wmma_f32(aA, bf, c0);
        c1 = wmma_f32(aB, bf, c1);
      }
#pragma unroll
      for (int j = 0; j < 8; ++j) {
        const int r = j + dRow;
        lds_ht[r * HID + n]        = relu(c0[j]);
        lds_ht[(16 + r) * HID + n] = relu(c1[j]);
      }
    }
    // 128 -> 3 head + residual: each of the 32 lanes owns one row.
    {
      const int rg = rowBase + lane;
      float a0 = bc[0], a1 = bc[1], a2 = bc[2];
      for (int i = 0; i < HID; ++i) {
        const float v = lds_ht[lane * HID + i];
        a0 += v * Wc[0 * HID + i];
        a1 += v * Wc[1 * HID + i];
        a2 += v * Wc[2 * HID + i];
      }
      float* o = out + ((size_t)rg * OUTW + t) * 3;
      o[0] = predicted[rg * 3 + 0] + a0;
      o[1] = predicted[rg * 3 + 1] + a1;
      o[2] = predicted[rg * 3 + 2] + a2;
    }
  }
}

// ---------------------------------------------------------------------------
extern "C" void kernel_launch(void* const* d_in, const int* in_sizes, int n_in,
                              void* d_out, int out_size, void* d_ws, size_t ws_size,
                              hipStream_t stream) {
  const float* partial   = (const float*)d_in[0];
  const float* predicted = (const float*)d_in[1];
  const float* W1 = (const float*)d_in[2];
  const float* b1 = (const float*)d_in[3];
  const float* W2 = (const float*)d_in[4];
  const float* b2 = (const float*)d_in[5];
  const float* Wd = (const float*)d_in[6];
  const float* bd = (const float*)d_in[7];
  const float* Wc = (const float*)d_in[8];
  const float* bc = (const float*)d_in[9];
  float* out = (float*)d_out;

  float* x = (float*)d_ws;                       // [B*N][8] = 2 MB scratch

  dim3 g1(NN / 256, BB);
  knn_seed_kernel<<<g1, 256, 0, stream>>>(partial, predicted, x);

  const int blocks = (BB * NN) / 128;            // 128 rows per block
  const size_t lds = (size_t)12 * 32 * HID * sizeof(float);   // 192 KB
  mlp_refine_kernel<<<blocks, 128, lds, stream>>>(
      x, predicted, W1, b1, W2, b2, Wd, bd, Wc, bc, out);
}